// GravesAttention_15994458210942
// MI455X (gfx1250) — compile-verified
//
#include <hip/hip_runtime.h>

typedef float v2f __attribute__((ext_vector_type(2)));
typedef float v8f __attribute__((ext_vector_type(8)));

#define COEF  0.3989422917366028f
#define EPS   1e-05f
#define ALIGN 0.05f

// Problem sizes (fixed by reference setup_inputs)
#define NB   1024   // batch
#define MEM  512    // mem_elem (MLP input dim)
#define HID  51     // hidden
#define K3   60     // 3*K
#define KK   20     // K
#define TT   500    // T
#define DD   256    // D

#define HIDP 64     // hidden padded for WMMA N/K
#define W1P_ELEMS (MEM * HIDP)    // 32768
#define W2P_ELEMS (HIDP * HIDP)   // 4096

// ---------------------------------------------------------------------------
// Kernel 0: zero-pad W1 (512x51 -> 512x64) and W2 (51x60 -> 64x64) into ws
//           so the WMMA loops are completely branch-free (no EXEC juggling).
// ---------------------------------------------------------------------------
__global__ __launch_bounds__(256) void k_pad(
    const float* __restrict__ W1, const float* __restrict__ W2,
    float* __restrict__ W1p, float* __restrict__ W2p)
{
    const int i = blockIdx.x * 256 + threadIdx.x;
    if (i < W1P_ELEMS) {
        const int r = i >> 6, c = i & 63;
        W1p[i] = (c < HID) ? W1[r * HID + c] : 0.0f;
    } else {
        const int j = i - W1P_ELEMS;
        if (j < W2P_ELEMS) {
            const int r = j >> 6, c = j & 63;
            W2p[j] = (r < HID && c < K3) ? W2[r * K3 + c] : 0.0f;
        }
    }
}

// ---------------------------------------------------------------------------
// Kernel 1: fused MLP  x(1024x512) -> h(1024x51,relu) -> gbk(1024x60)
//           + per-row softmax/exp/mu update.
// One wave32 per 16-row M tile. f32 WMMA 16x16x4, N padded to 64.
// Branch-free inner loops thanks to pre-padded weights.
// ---------------------------------------------------------------------------
__global__ __launch_bounds__(32) void k_mlp(
    const float* __restrict__ X,    // (1024, 512)  = C flattened
    const float* __restrict__ MU0,  // (1024, 20)
    const float* __restrict__ W1p,  // (512, 64) padded
    const float* __restrict__ B1,   // (51)
    const float* __restrict__ W2p,  // (64, 64) padded
    const float* __restrict__ B2,   // (60)
    float* __restrict__ mu_out,     // (1024, 20)  -> d_out region 2
    float* __restrict__ gsm)        // (1024, 3, 20) ws: g, sig, mu
{
    __shared__ float sh[16][65];    // padded: column reads hit distinct banks

    const int lane = threadIdx.x;
    const int mloc = lane & 15;         // M (A rows) / N (B cols) index
    const int kh   = (lane >> 4) << 1;  // K sub-offset: 0 or 2
    const int rb   = (lane >> 4) << 3;  // C/D row base: 0 or 8
    const int row  = blockIdx.x * 16 + mloc;

    // ---------------- GEMM1: h = relu(X @ W1 + b1), N padded to 64 ---------
    v8f acc[4] = {};
    for (int k = 0; k < MEM; k += 4) {
        v2f a;
        a.x = X[row * MEM + k + kh];
        a.y = X[row * MEM + k + kh + 1];
#pragma unroll
        for (int nt = 0; nt < 4; ++nt) {
            const int n = nt * 16 + mloc;
            v2f b;
            b.x = W1p[(k + kh)     * HIDP + n];
            b.y = W1p[(k + kh + 1) * HIDP + n];
            acc[nt] = __builtin_amdgcn_wmma_f32_16x16x4_f32(
                false, a, false, b, (short)0, acc[nt], false, false);
        }
    }
    // bias + relu, stage h in LDS (convert D layout -> addressable rows)
#pragma unroll
    for (int nt = 0; nt < 4; ++nt) {
        const int n = nt * 16 + mloc;
        const float bv = (n < HID) ? B1[n] : 0.0f;
#pragma unroll
        for (int r = 0; r < 8; ++r) {
            float h = acc[nt][r] + bv;
            h = (h > 0.0f) ? h : 0.0f;
            if (n >= HID) h = 0.0f;     // keep pad columns exactly zero
            sh[rb + r][n] = h;
        }
    }
    __syncthreads();

    // ---------------- GEMM2: gbk = h @ W2 + b2, K padded to 64 -------------
    v8f acc2[4] = {};
#pragma unroll
    for (int k = 0; k < HIDP; k += 4) {
        v2f a;
        a.x = sh[mloc][k + kh];
        a.y = sh[mloc][k + kh + 1];
#pragma unroll
        for (int nt = 0; nt < 4; ++nt) {
            const int n = nt * 16 + mloc;
            v2f b;
            b.x = W2p[(k + kh)     * HIDP + n];
            b.y = W2p[(k + kh + 1) * HIDP + n];
            acc2[nt] = __builtin_amdgcn_wmma_f32_16x16x4_f32(
                false, a, false, b, (short)0, acc2[nt], false, false);
        }
    }
    __syncthreads();
#pragma unroll
    for (int nt = 0; nt < 4; ++nt) {
        const int n = nt * 16 + mloc;
        const float bv = (n < K3) ? B2[n] : 0.0f;
#pragma unroll
        for (int r = 0; r < 8; ++r)
            sh[rb + r][n] = acc2[nt][r] + bv;
    }
    __syncthreads();

    // ---------------- per-row softmax / exp / mu (lanes 0..15 own a row) ---
    if (lane < 16) {
        const int r = lane;
        const int grow = blockIdx.x * 16 + r;
        float mx = -1e30f;
#pragma unroll
        for (int k = 0; k < KK; ++k) mx = fmaxf(mx, sh[r][k]);
        float e[KK];
        float s = 0.0f;
#pragma unroll
        for (int k = 0; k < KK; ++k) { e[k] = __expf(sh[r][k] - mx); s += e[k]; }
        const float inv = 1.0f / s;
#pragma unroll
        for (int k = 0; k < KK; ++k) {
            const float g  = e[k] * inv + EPS;
            const float sg = __expf(sh[r][KK + k]) + EPS;
            const float mu = MU0[grow * KK + k] + ALIGN * __expf(sh[r][2 * KK + k]);
            gsm[grow * 60 +      k] = g;
            gsm[grow * 60 + 20 + k] = sg;
            gsm[grow * 60 + 40 + k] = mu;
            mu_out[grow * KK + k]   = mu;
        }
    }
}

// ---------------------------------------------------------------------------
// Kernel 2: per-batch alpha (Gaussian mixture over T) + weighted reduction
//           c[d] = sum_t alpha[t] * ctx[b,t,d].  This is the 512 MB stream:
//           float4 (b128) coalesced loads, 4-way T-partitioned accumulators.
// ---------------------------------------------------------------------------
__global__ __launch_bounds__(256) void k_attn(
    const float* __restrict__ ctx,   // (1024, 500, 256)
    const float* __restrict__ gsm,   // (1024, 3, 20)
    float* __restrict__ c_out,       // (1024, 256)
    float* __restrict__ alpha_out)   // (1024, 500)
{
    __shared__ float g[KK], sg[KK], mu[KK];
    __shared__ float alpha[TT];
    __shared__ float4 part[4][64];

    const int b   = blockIdx.x;
    const int tid = threadIdx.x;

    if (tid < 60) {
        const float v = gsm[b * 60 + tid];
        if (tid < 20)      g[tid]       = v;
        else if (tid < 40) sg[tid - 20] = v;
        else               mu[tid - 40] = v;
    }
    __syncthreads();

    // Phase A: alpha[t] = COEF * sum_k g[k] * exp(-0.5*sig[k]*(mu[k]-t)^2)
    for (int t = tid; t < TT; t += 256) {
        const float ft = (float)t;
        float s = 0.0f;
#pragma unroll
        for (int k = 0; k < KK; ++k) {
            const float d = mu[k] - ft;
            s += g[k] * __expf(-0.5f * sg[k] * d * d);
        }
        const float a = COEF * s;
        alpha[t] = a;
        alpha_out[(size_t)b * TT + t] = a;
    }
    __syncthreads();

    // Phase B: c[d] = sum_t alpha[t] * ctx[t, d]   (bandwidth phase)
    const int d4  = tid & 63;   // which float4 of the 256-wide D row
    const int grp = tid >> 6;   // 4-way partition of T
    float4 acc = make_float4(0.f, 0.f, 0.f, 0.f);
    const float4* cb = (const float4*)(ctx + (size_t)b * TT * DD);
    for (int t = grp; t < TT; t += 4) {
        const float a  = alpha[t];
        const float4 v = cb[t * 64 + d4];
        acc.x += a * v.x; acc.y += a * v.y; acc.z += a * v.z; acc.w += a * v.w;
    }
    part[grp][d4] = acc;
    __syncthreads();

    if (tid < 64) {
        float4 r = make_float4(0.f, 0.f, 0.f, 0.f);
#pragma unroll
        for (int gi = 0; gi < 4; ++gi) {
            const float4 p = part[gi][tid];
            r.x += p.x; r.y += p.y; r.z += p.z; r.w += p.w;
        }
        ((float4*)(c_out + (size_t)b * DD))[tid] = r;
    }
}

// ---------------------------------------------------------------------------
extern "C" void kernel_launch(void* const* d_in, const int* in_sizes, int n_in,
                              void* d_out, int out_size, void* d_ws, size_t ws_size,
                              hipStream_t stream) {
    (void)in_sizes; (void)n_in; (void)out_size; (void)ws_size;
    const float* C   = (const float*)d_in[0];   // (1024, 2, 256)
    const float* ctx = (const float*)d_in[1];   // (1024, 500, 256)
    const float* mu0 = (const float*)d_in[2];   // (1024, 20)
    const float* W1  = (const float*)d_in[3];   // (512, 51)
    const float* b1  = (const float*)d_in[4];   // (51)
    const float* W2  = (const float*)d_in[5];   // (51, 60)
    const float* b2  = (const float*)d_in[6];   // (60)

    float* out     = (float*)d_out;
    float* c_t     = out;                         // 1024*256
    float* mu_t    = out + NB * DD;               // 1024*20
    float* alpha_t = out + NB * DD + NB * KK;     // 1024*500

    float* gsm = (float*)d_ws;                    // 1024*3*20 floats
    float* W1p = gsm + NB * 60;                   // 512*64 floats
    float* W2p = W1p + W1P_ELEMS;                 // 64*64 floats

    const int pad_threads = W1P_ELEMS + W2P_ELEMS;          // 36864
    k_pad <<<(pad_threads + 255) / 256, 256, 0, stream>>>(W1, W2, W1p, W2p);
    k_mlp <<<NB / 16, 32,  0, stream>>>(C, mu0, W1p, b1, W2p, b2, mu_t, gsm);
    k_attn<<<NB,      256, 0, stream>>>(ctx, gsm, c_t, alpha_t);
}